// GCN_80625126080958
// MI455X (gfx1250) — compile-verified
//
#include <hip/hip_runtime.h>
#include <hip/hip_bf16.h>
#include <math.h>

#define N_NODES   100000
#define N_EDGES   3200000

typedef float v2f __attribute__((ext_vector_type(2)));
typedef float v8f __attribute__((ext_vector_type(8)));

// ---------------------------------------------------------------------------
// deg[i] = 1.0 (self loop); then atomically count occurrences in col[]
// ---------------------------------------------------------------------------
__global__ void deg_init_kernel(float* __restrict__ deg, int n) {
    int i = blockIdx.x * blockDim.x + threadIdx.x;
    if (i < n) deg[i] = 1.0f;  // self-loop contribution
}

__global__ void deg_accum_kernel(const int* __restrict__ col,
                                 float* __restrict__ deg, int ne) {
    int e = blockIdx.x * blockDim.x + threadIdx.x;
    if (e >= ne) return;
    __hip_atomic_fetch_add(&deg[col[e]], 1.0f,
                           __ATOMIC_RELAXED, __HIP_MEMORY_SCOPE_AGENT);
}

__global__ void dinv_kernel(float* __restrict__ dinv, int n) {
    int i = blockIdx.x * blockDim.x + threadIdx.x;
    if (i < n) dinv[i] = rsqrtf(dinv[i]);  // deg >= 1 always
}

__global__ void norm_kernel(const int* __restrict__ row,
                            const int* __restrict__ col,
                            const float* __restrict__ dinv,
                            float* __restrict__ norm, int ne) {
    int e = blockIdx.x * blockDim.x + threadIdx.x;
    if (e >= ne) return;
    norm[e] = dinv[row[e]] * dinv[col[e]];
}

// ---------------------------------------------------------------------------
// Layer-1 GEMM: hlin[N,4] = x[N,128] @ W1[128,4] via V_WMMA_F32_16X16X4_F32.
// One wave computes a 16-row tile; W1 zero-padded to 128x16 in LDS.
// A 16x4 frag: lane half h, VGPR v -> K = k + v + 2h, M = lane&15.
// B 4x16 frag: same K indexing, N = lane&15.
// C 16x16:     VGPR v, half h -> M = v + 8h, N = lane&15.
// ---------------------------------------------------------------------------
__global__ void __launch_bounds__(256)
gcn_wmma_xw1_kernel(const float* __restrict__ x,
                    const float* __restrict__ W1,   // [128,4]
                    float* __restrict__ hlin,       // [N,4]
                    int n) {
    __shared__ float Wp[128 * 16];                  // 8 KB: W1 padded to 128x16
    const int t = threadIdx.x;
    #pragma unroll
    for (int j = 0; j < 8; ++j) {                   // 256 thr * 8 = 2048 entries
        int idx = t * 8 + j;
        int kr  = idx >> 4;
        int nc  = idx & 15;
        Wp[idx] = (nc < 4) ? W1[kr * 4 + nc] : 0.0f;
    }
    __syncthreads();

    const int wave  = t >> 5;                        // 8 waves / block
    const int lane  = t & 31;
    const int tile  = blockIdx.x * 8 + wave;
    const int nTiles = n >> 4;                       // 100000/16 = 6250 exactly
    if (tile >= nTiles) return;                      // wave-uniform exit

    const int base = tile * 16;
    const int hh   = lane >> 4;                      // lane half: 0 or 1
    const int nl   = lane & 15;
    const int arow = base + nl;

    v8f c = {};
    #pragma unroll 4
    for (int k = 0; k < 128; k += 4) {
        const int kk = k + 2 * hh;
        v2f a = *reinterpret_cast<const v2f*>(x + (size_t)arow * 128 + kk);
        v2f b;
        b.x = Wp[(kk + 0) * 16 + nl];
        b.y = Wp[(kk + 1) * 16 + nl];
        c = __builtin_amdgcn_wmma_f32_16x16x4_f32(
                /*neg_a=*/false, a, /*neg_b=*/false, b,
                /*c_mod=*/(short)0, c, /*reuse_a=*/false, /*reuse_b=*/false);
    }

    if (nl < 4) {                                    // only 4 valid output cols
        #pragma unroll
        for (int v = 0; v < 8; ++v) {
            int m = base + v + 8 * hh;
            hlin[(size_t)m * 4 + nl] = c[v];
        }
    }
}

// ---------------------------------------------------------------------------
// Tiny dense transforms for layers 2/3 (IN,OUT <= 4): one thread per node.
// ---------------------------------------------------------------------------
template <int IN, int OUT>
__global__ void dense_small_kernel(const float* __restrict__ h,
                                   const float* __restrict__ W,  // [IN,OUT]
                                   float* __restrict__ out, int n) {
    int i = blockIdx.x * blockDim.x + threadIdx.x;
    if (i >= n) return;
    float hv[IN];
    #pragma unroll
    for (int k = 0; k < IN; ++k) hv[k] = h[(size_t)i * IN + k];
    #pragma unroll
    for (int o = 0; o < OUT; ++o) {
        float s = 0.0f;
        #pragma unroll
        for (int k = 0; k < IN; ++k) s = fmaf(hv[k], W[k * OUT + o], s);
        out[(size_t)i * OUT + o] = s;
    }
}

// ---------------------------------------------------------------------------
// Per-edge gather * norm -> scatter-add (hardware fp32 atomics in L2).
// ---------------------------------------------------------------------------
template <int OUT>
__global__ void edge_scatter_kernel(const int* __restrict__ row,
                                    const int* __restrict__ col,
                                    const float* __restrict__ norm,
                                    const float* __restrict__ hlin,
                                    float* __restrict__ agg, int ne) {
    int e = blockIdx.x * blockDim.x + threadIdx.x;
    if (e >= ne) return;
    const int r = row[e];
    const int c = col[e];
    const float nm = norm[e];
    float msg[OUT];
    if (OUT == 4) {
        float4 v = reinterpret_cast<const float4*>(hlin)[c];
        msg[0] = v.x; msg[1] = v.y; msg[2] = v.z; msg[3] = v.w;
    } else if (OUT == 2) {
        float2 v = reinterpret_cast<const float2*>(hlin)[c];
        msg[0] = v.x; msg[1] = v.y;
    } else {
        #pragma unroll
        for (int f = 0; f < OUT; ++f) msg[f] = hlin[(size_t)c * OUT + f];
    }
    #pragma unroll
    for (int f = 0; f < OUT; ++f) {
        __hip_atomic_fetch_add(&agg[(size_t)r * OUT + f], msg[f] * nm,
                               __ATOMIC_RELAXED, __HIP_MEMORY_SCOPE_AGENT);
    }
}

// ---------------------------------------------------------------------------
// Epilogue: add self-loop term (dinv^2 * hlin), bias, tanh.
// ---------------------------------------------------------------------------
template <int OUT>
__global__ void finalize_kernel(const float* __restrict__ agg,
                                const float* __restrict__ hlin,
                                const float* __restrict__ dinv,
                                const float* __restrict__ b,
                                float* __restrict__ out, int n) {
    int i = blockIdx.x * blockDim.x + threadIdx.x;
    if (i >= n) return;
    const float d2 = dinv[i] * dinv[i];
    #pragma unroll
    for (int f = 0; f < OUT; ++f) {
        float v = agg[(size_t)i * OUT + f]
                + hlin[(size_t)i * OUT + f] * d2 + b[f];
        out[(size_t)i * OUT + f] = tanhf(v);
    }
}

// ---------------------------------------------------------------------------
// Classifier: out[N,4] = h[N,2] @ Wc[2,4] + bc
// ---------------------------------------------------------------------------
__global__ void classify_kernel(const float* __restrict__ h,
                                const float* __restrict__ Wc,
                                const float* __restrict__ bc,
                                float* __restrict__ out, int n) {
    int i = blockIdx.x * blockDim.x + threadIdx.x;
    if (i >= n) return;
    const float h0 = h[(size_t)i * 2 + 0];
    const float h1 = h[(size_t)i * 2 + 1];
    #pragma unroll
    for (int c = 0; c < 4; ++c) {
        out[(size_t)i * 4 + c] = fmaf(h0, Wc[c], fmaf(h1, Wc[4 + c], bc[c]));
    }
}

// ---------------------------------------------------------------------------
extern "C" void kernel_launch(void* const* d_in, const int* in_sizes, int n_in,
                              void* d_out, int out_size, void* d_ws, size_t ws_size,
                              hipStream_t stream) {
    const float* x  = (const float*)d_in[0];
    const int*   ei = (const int*)d_in[1];          // [2, E] int32 (x64 disabled)
    const float* W1 = (const float*)d_in[2];
    const float* b1 = (const float*)d_in[3];
    const float* W2 = (const float*)d_in[4];
    const float* b2 = (const float*)d_in[5];
    const float* W3 = (const float*)d_in[6];
    const float* b3 = (const float*)d_in[7];
    const float* Wc = (const float*)d_in[8];
    const float* bc = (const float*)d_in[9];

    const int N = N_NODES;
    const int E = N_EDGES;
    const int* row = ei;          // destinations
    const int* col = ei + E;      // sources

    // workspace layout (floats)
    float* ws   = (float*)d_ws;
    float* dinv = ws;             // N     (holds deg, then rsqrt(deg))
    float* norm = dinv + N;       // E
    float* hlin = norm + E;       // 4N
    float* agg  = hlin + 4 * N;   // 4N
    float* hA   = agg  + 4 * N;   // 4N
    float* hB   = hA   + 4 * N;   // 4N

    float* outC = (float*)d_out;          // [N,4]
    float* outH = outC + (size_t)4 * N;   // [N,2] -- layer-3 output lives here

    const int BT = 256;
    const int gN = (N + BT - 1) / BT;
    const int gE = (E + BT - 1) / BT;

    // --- degree / normalization --------------------------------------------
    deg_init_kernel<<<gN, BT, 0, stream>>>(dinv, N);
    deg_accum_kernel<<<gE, BT, 0, stream>>>(col, dinv, E);
    dinv_kernel<<<gN, BT, 0, stream>>>(dinv, N);
    norm_kernel<<<gE, BT, 0, stream>>>(row, col, dinv, norm, E);

    // --- layer 1: x@W1 (WMMA), scatter, tanh -------------------------------
    {
        const int tiles  = N / 16;                 // 6250
        const int blocks = (tiles + 7) / 8;        // 8 waves/block
        gcn_wmma_xw1_kernel<<<blocks, 256, 0, stream>>>(x, W1, hlin, N);
        hipMemsetAsync(agg, 0, (size_t)4 * N * sizeof(float), stream);
        edge_scatter_kernel<4><<<gE, BT, 0, stream>>>(row, col, norm, hlin, agg, E);
        finalize_kernel<4><<<gN, BT, 0, stream>>>(agg, hlin, dinv, b1, hA, N);
    }
    // --- layer 2 ------------------------------------------------------------
    {
        dense_small_kernel<4, 4><<<gN, BT, 0, stream>>>(hA, W2, hlin, N);
        hipMemsetAsync(agg, 0, (size_t)4 * N * sizeof(float), stream);
        edge_scatter_kernel<4><<<gE, BT, 0, stream>>>(row, col, norm, hlin, agg, E);
        finalize_kernel<4><<<gN, BT, 0, stream>>>(agg, hlin, dinv, b2, hB, N);
    }
    // --- layer 3 (writes h directly into the output buffer) ----------------
    {
        dense_small_kernel<4, 2><<<gN, BT, 0, stream>>>(hB, W3, hlin, N);
        hipMemsetAsync(agg, 0, (size_t)2 * N * sizeof(float), stream);
        edge_scatter_kernel<2><<<gE, BT, 0, stream>>>(row, col, norm, hlin, agg, E);
        finalize_kernel<2><<<gN, BT, 0, stream>>>(agg, hlin, dinv, b3, outH, N);
    }
    // --- classifier ---------------------------------------------------------
    classify_kernel<<<gN, BT, 0, stream>>>(outH, Wc, bc, outC, N);
}